// VariationalGCNEncoder_67774583931485
// MI455X (gfx1250) — compile-verified
//
#include <hip/hip_runtime.h>

typedef float v2f __attribute__((ext_vector_type(2)));
typedef float v8f __attribute__((ext_vector_type(8)));
typedef unsigned int u32x4 __attribute__((ext_vector_type(4)));
typedef int i32x4 __attribute__((ext_vector_type(4)));
typedef int i32x8 __attribute__((ext_vector_type(8)));

#define CDIM 128          // C_IN == C_OUT == 128
#define NT   (CDIM / 16)  // 8 column tiles of 16

// ---------------------------------------------------------------------------
// Zero-fill (workspace agg buffer); float4 grid-stride.
// ---------------------------------------------------------------------------
__global__ void zero_f4_kernel(float4* __restrict__ p, long n4) {
    long i = (long)blockIdx.x * blockDim.x + threadIdx.x;
    long stride = (long)gridDim.x * blockDim.x;
    float4 z = make_float4(0.f, 0.f, 0.f, 0.f);
    for (; i < n4; i += stride) p[i] = z;
}

// ---------------------------------------------------------------------------
// Edge scatter: one wave (32 lanes) per edge. Lane l handles channels
// 4l..4l+3 (float4) -> 512B coalesced gather of x[src], then 4 f32 atomics
// into agg[dst]. edge_index is (2,E) flat: src at [0,E), dst at [E,2E).
// ---------------------------------------------------------------------------
__global__ void edge_scatter_kernel(const float* __restrict__ x,
                                    const int* __restrict__ edge_index,
                                    const float* __restrict__ edge_weight,
                                    float* __restrict__ agg, int E) {
    int gtid = blockIdx.x * blockDim.x + threadIdx.x;
    int e    = gtid >> 5;
    int lane = threadIdx.x & 31;
    if (e >= E) return;

    int   src = edge_index[e];
    int   dst = edge_index[E + e];
    float w   = edge_weight[e];

    const float4* xrow = (const float4*)(x + (size_t)src * CDIM);
    float4 v = xrow[lane];

    float* arow = agg + (size_t)dst * CDIM + lane * 4;
    atomicAdd(arow + 0, v.x * w);
    atomicAdd(arow + 1, v.y * w);
    atomicAdd(arow + 2, v.z * w);
    atomicAdd(arow + 3, v.w * w);
}

// ---------------------------------------------------------------------------
// TDM stage: one wave issues a single tensor_load_to_lds moving a 64KB
// weight matrix (16384 f32, contiguous) into LDS, then waits on TENSORcnt.
// D# per CDNA5 ISA ch.8:
//   group0: [count=1 | lds_addr | global_addr[56:0] | type=2]
//   group1: [data_size=2(4B) | tensor_dim0=16384 | tensor_dim1=1 |
//            tile_dim0=16384 | tile_dim1/2=0 | dim0_stride=16384]
//   group2/3: zero (tensor <= 2D)
// ---------------------------------------------------------------------------
__device__ __forceinline__ void tdm_load_weights(const float* gsrc, float* lds_dst) {
    unsigned long long ga = (unsigned long long)(uintptr_t)gsrc;
    unsigned lds = (unsigned)(uintptr_t)(void*)lds_dst;  // low 32b = LDS offset

    u32x4 g0;
    g0.x = 1u;                                    // count=1 valid descriptor
    g0.y = lds;                                   // lds_addr (bytes)
    g0.z = (unsigned)(ga & 0xFFFFFFFFu);          // global_addr[31:0]
    g0.w = (unsigned)((ga >> 32) & 0x1FFFFFFu)    // global_addr[56:32]
           | (2u << 30);                          // type=2 ("image")

    i32x8 g1;
    g1[0] = 0x00020000;   // workgroup_mask=0, data_size=2 (4 bytes)
    g1[1] = 0x40000000;   // tensor_dim0 = 16384 -> bits[63:48]
    g1[2] = 0x00010000;   // tensor_dim1 = 1     -> bits[111:80]
    g1[3] = 0x40000000;   // tile_dim0  = 16384 -> bits[127:112]
    g1[4] = 0;            // tile_dim1 = 0 (unused), tile_dim2 = 0
    g1[5] = 16384;        // tensor_dim0_stride[31:0]
    g1[6] = 0;            // stride hi, dim1_stride lo
    g1[7] = 0;

    i32x4 gz = {0, 0, 0, 0};

#if __clang_major__ >= 23
    i32x8 gz8 = {0, 0, 0, 0, 0, 0, 0, 0};
    __builtin_amdgcn_tensor_load_to_lds(g0, g1, gz, gz, gz8, 0);
#else
    __builtin_amdgcn_tensor_load_to_lds(g0, g1, gz, gz, 0);
#endif
    __builtin_amdgcn_s_wait_tensorcnt(0);
}

// ---------------------------------------------------------------------------
// Fused GEMM + bias + ReLU using V_WMMA_F32_16X16X4_F32 (full f32 precision).
//   out[M,128] = relu( sum_p A_p[M,128] @ W_p[128,128]  + bias )
// Block: 256 threads = 8 waves; each wave owns a 16-row strip and all 8
// 16-wide column tiles (8x v8f accumulators). Weights staged in 64KB LDS per
// pass via the Tensor Data Mover and shared by all waves.
//
// f32 WMMA fragment layout (ISA 7.12.2), wave32:
//   A 16x4 : lane l -> row m = l&15; VGPR j -> k = 2*(l>>4) + j
//   B 4x16 : lane l -> col n = l&15; VGPR j -> k = 2*(l>>4) + j
//   C 16x16: lane l -> col n = l&15; VGPR i -> row m = i + 8*(l>>4)
// ---------------------------------------------------------------------------
template <int NPASS>
__global__ __launch_bounds__(256) void gemm_bias_relu_kernel(
        const float* __restrict__ A0, const float* __restrict__ A1,
        const float* __restrict__ W0, const float* __restrict__ W1,
        const float* __restrict__ bias, float* __restrict__ out, int M) {
    __shared__ float sw[CDIM * CDIM];  // 64 KB: current pass's weight matrix

    const int tid   = threadIdx.x;
    const int wave  = tid >> 5;
    const int lane  = tid & 31;
    const int m0    = (blockIdx.x * 8 + wave) * 16;  // wave's row-strip base
    const int sub   = lane & 15;                     // A row / B,C column
    const int khalf = lane >> 4;                     // K-half select

    v8f acc[NT] = {};  // 8 tiles x 8 f32 accumulators

    const float* Alist[2] = {A0, A1};
    const float* Wlist[2] = {W0, W1};

    for (int p = 0; p < NPASS; ++p) {
        __syncthreads();  // previous pass finished reading LDS
        if (wave == 0) {  // one wave drives the TDM; TENSORcnt tracks it
            tdm_load_weights(Wlist[p], sw);
        }
        __syncthreads();  // weights visible to all 8 waves

        if (m0 < M) {  // wave-uniform guard: EXEC stays all-ones for WMMA
            const float* arow = Alist[p] + (size_t)(m0 + sub) * CDIM;
            for (int kb = 0; kb < CDIM; kb += 4) {
                const int k0 = kb + 2 * khalf;
                v2f a;
                a.x = arow[k0 + 0];
                a.y = arow[k0 + 1];
#pragma unroll
                for (int t = 0; t < NT; ++t) {
                    v2f b;
                    b.x = sw[(k0 + 0) * CDIM + t * 16 + sub];
                    b.y = sw[(k0 + 1) * CDIM + t * 16 + sub];
                    acc[t] = __builtin_amdgcn_wmma_f32_16x16x4_f32(
                        /*neg_a=*/false, a, /*neg_b=*/false, b,
                        /*c_mod=*/(short)0, acc[t],
                        /*reuse_a=*/false, /*reuse_b=*/false);
                }
            }
        }
    }

    if (m0 >= M) return;

    // Bias + ReLU + store. Row of acc[t][i] is m0 + i + 8*khalf, col t*16+sub.
#pragma unroll
    for (int t = 0; t < NT; ++t) {
        float bv = bias[t * 16 + sub];
#pragma unroll
        for (int i = 0; i < 8; ++i) {
            float v = acc[t][i] + bv;
            v = v > 0.f ? v : 0.f;
            out[(size_t)(m0 + i + 8 * khalf) * CDIM + t * 16 + sub] = v;
        }
    }
}

// ---------------------------------------------------------------------------
// Launch: zero agg -> edge scatter -> GEMM1 (agg@W_rel + x@W_root) ->
//         GEMM2 (mu) -> GEMM2 (log_std).
// ---------------------------------------------------------------------------
extern "C" void kernel_launch(void* const* d_in, const int* in_sizes, int n_in,
                              void* d_out, int out_size, void* d_ws, size_t ws_size,
                              hipStream_t stream) {
    const float* x      = (const float*)d_in[0];
    const int*   eindex = (const int*)d_in[1];
    const float* eweight= (const float*)d_in[2];
    const float* W_rel  = (const float*)d_in[3];
    const float* b_rel  = (const float*)d_in[4];
    const float* W_root = (const float*)d_in[5];
    const float* W_mu   = (const float*)d_in[6];
    const float* b_mu   = (const float*)d_in[7];
    const float* W_std  = (const float*)d_in[8];
    const float* b_std  = (const float*)d_in[9];

    const int N = in_sizes[0] / CDIM;   // 50000
    const int E = in_sizes[2];          // 800000

    float* agg    = (float*)d_ws;                       // N*128 f32
    float* hidden = agg + (size_t)N * CDIM;             // N*128 f32
    float* mu_out = (float*)d_out;                      // N*128 f32
    float* ls_out = mu_out + (size_t)N * CDIM;          // N*128 f32

    // 1) zero the aggregation buffer
    {
        long n4 = ((long)N * CDIM) / 4;
        int blocks = (int)((n4 + 255) / 256);
        zero_f4_kernel<<<blocks, 256, 0, stream>>>((float4*)agg, n4);
    }
    // 2) edge gather * weight -> atomic scatter-add into agg
    {
        int waves_per_block = 256 / 32;
        int blocks = (E + waves_per_block - 1) / waves_per_block;
        edge_scatter_kernel<<<blocks, 256, 0, stream>>>(x, eindex, eweight, agg, E);
    }
    // 3) hidden = relu(agg@W_rel + x@W_root + b_rel)
    {
        int strips = (N + 15) / 16;             // 3125
        int blocks = (strips + 7) / 8;          // 391
        gemm_bias_relu_kernel<2><<<blocks, 256, 0, stream>>>(
            agg, x, W_rel, W_root, b_rel, hidden, N);
    }
    // 4) mu = relu(hidden@W_mu + b_mu); log_std = relu(hidden@W_std + b_std)
    {
        int strips = (N + 15) / 16;
        int blocks = (strips + 7) / 8;
        gemm_bias_relu_kernel<1><<<blocks, 256, 0, stream>>>(
            hidden, nullptr, W_mu, nullptr, b_mu, mu_out, N);
        gemm_bias_relu_kernel<1><<<blocks, 256, 0, stream>>>(
            hidden, nullptr, W_std, nullptr, b_std, ls_out, N);
    }
}